// ProteinGCN_21560735826433
// MI455X (gfx1250) — compile-verified
//
#include <hip/hip_runtime.h>
#include <hip/hip_bf16.h>
#include <math.h>

typedef __attribute__((ext_vector_type(2))) float v2f;
typedef __attribute__((ext_vector_type(8))) float v8f;

#define H1 64
#define H2 128

// ---- Phase 0: deg init (self-loop contributes 1 to every node) ----
__global__ void gcn_init_deg(float* __restrict__ deg, int N) {
    int i = blockIdx.x * blockDim.x + threadIdx.x;
    if (i < N) deg[i] = 1.0f;
}

// ---- Phase 1: degree over edge targets ----
__global__ void gcn_deg(const int* __restrict__ col, float* __restrict__ deg, int E) {
    int i = blockIdx.x * blockDim.x + threadIdx.x;
    if (i < E) atomicAdd(&deg[col[i]], 1.0f);
}

// ---- Phase 2a: dinv = rsqrt(deg); s initialized with self-loop term x[i]*dinv^2 ----
__global__ void gcn_dinv_sinit(const float* __restrict__ x, float* __restrict__ dv,
                               float* __restrict__ s, int N) {
    int i = blockIdx.x * blockDim.x + threadIdx.x;
    if (i < N) {
        float d = dv[i];
        float di = d > 0.0f ? rsqrtf(d) : 0.0f;
        dv[i] = di;
        s[i] = x[i] * di * di;
    }
}

// ---- Phase 2b: scalar layer-1 aggregation: s[c] += x[r]*dinv[r]*dinv[c] ----
__global__ void gcn_sagg(const float* __restrict__ x, const int* __restrict__ row,
                         const int* __restrict__ col, const float* __restrict__ dinv,
                         float* __restrict__ s, int E) {
    int tid = blockIdx.x * blockDim.x + threadIdx.x;
    int stride = gridDim.x * blockDim.x;
    for (int e = tid; e < E; e += stride) {
        int r = row[e], c = col[e];
        atomicAdd(&s[c], x[r] * dinv[r] * dinv[c]);
    }
}

// ---- Phase 3a: agg init with layer-2 self-loop: dinv[i]^2 * relu(s[i]*W1[j]+b1[j]) ----
__global__ void gcn_agg_init(const float* __restrict__ s, const float* __restrict__ dinv,
                             const float* __restrict__ W1, const float* __restrict__ b1,
                             float* __restrict__ agg, int N) {
    int t = blockIdx.x * blockDim.x + threadIdx.x;
    int i = t >> 6;
    int j = t & 63;
    if (i < N) {
        float di = dinv[i];
        float h = fmaxf(fmaf(s[i], W1[j], b1[j]), 0.0f);
        agg[(size_t)i * H1 + j] = di * di * h;
    }
}

// ---- Phase 3b: main edge pass. 64 threads per edge; each wave's 32 lanes share one
//      edge, so the gathers are wave-uniform (readfirstlane -> scalar loads), and the
//      scatter is 32 consecutive f32 atomics (agg is L2-resident: 25.6MB << 192MB). ----
__global__ void gcn_edge_agg(const int* __restrict__ row, const int* __restrict__ col,
                             const float* __restrict__ s, const float* __restrict__ dinv,
                             const float* __restrict__ W1, const float* __restrict__ b1,
                             float* __restrict__ agg, int E) {
    int tid = blockIdx.x * blockDim.x + threadIdx.x;
    int j = tid & 63;                        // fixed output column per thread
    int slot = tid >> 6;
    int nslots = (gridDim.x * blockDim.x) >> 6;
    float w = W1[j];
    float bb = b1[j];
    for (int e = slot; e < E; e += nslots) {
        int eu = __builtin_amdgcn_readfirstlane(e);   // wave-uniform edge id
        int r = row[eu];
        int c = col[eu];
        float norm = dinv[r] * dinv[c];
        float h = fmaxf(fmaf(s[r], w, bb), 0.0f);
        atomicAdd(&agg[(size_t)c * H1 + j], norm * h);
    }
}

// ---- Phase 4: out = agg[N,64] @ W2[64,128] + b2 via V_WMMA_F32_16X16X4_F32.
//      One wave per 16x16 output tile; A/B fragments preloaded, 16 chained WMMAs.
//      Out-of-range A rows are CLAMPED (their C rows are never written), so there
//      is no per-element predication around the loads. ----
__global__ void gcn_gemm_wmma(const float* __restrict__ agg, const float* __restrict__ W2,
                              const float* __restrict__ b2, float* __restrict__ out, int N) {
    int lane = threadIdx.x & 31;
    int w = (blockIdx.x * blockDim.x + threadIdx.x) >> 5;
    int rowTile = w >> 3;            // 16 nodes per tile
    int colTile = w & 7;             // 8 tiles of 16 columns
    if (rowTile * 16 >= N) return;   // wave-uniform exit

    int m = lane & 15;               // M (for A) / N (for B,C)
    int khalf = lane >> 4;           // 0: K pair {0,1}, 1: K pair {2,3}
    int node = rowTile * 16 + m;
    if (node >= N) node = N - 1;     // clamp: row m only affects output row m
    const float* arow = agg + (size_t)node * H1;
    int cbase = colTile * 16 + m;

    // Preload fragments. A 16x4 f32: lanes0-15 M,K={0,1}; lanes16-31 M,K={2,3}.
    // B 4x16 f32 mirrored over N. W2 is [K=64, N=128] row-major.
    v2f af[16], bf[16];
    #pragma unroll
    for (int kk = 0; kk < 16; ++kk) {
        int kb = kk * 4 + khalf * 2;
        af[kk] = *(const v2f*)(arow + kb);              // global_load_b64
        bf[kk][0] = W2[(size_t)kb * H2 + cbase];
        bf[kk][1] = W2[(size_t)(kb + 1) * H2 + cbase];
    }

    v8f c = {};
    #pragma unroll
    for (int kk = 0; kk < 16; ++kk) {
        c = __builtin_amdgcn_wmma_f32_16x16x4_f32(false, af[kk], false, bf[kk],
                                                  (short)0, c, false, false);
    }

    // C/D 16x16 f32: VGPR i -> M=i (lanes0-15) / M=i+8 (lanes16-31), N = lane%16
    float bias = b2[cbase];
    int rbase = rowTile * 16 + khalf * 8;
    #pragma unroll
    for (int i = 0; i < 8; ++i) {
        int rr = rbase + i;
        if (rr < N) out[(size_t)rr * H2 + cbase] = c[i] + bias;
    }
}

// ---- Phase 5: row-wise log_softmax over 128 classes. One wave per row;
//      each lane owns 4 contiguous classes -> single b128 load/store per lane. ----
__global__ void gcn_log_softmax(float* __restrict__ out, int N) {
    int lane = threadIdx.x & 31;
    int r = (blockIdx.x * blockDim.x + threadIdx.x) >> 5;
    if (r >= N) return;
    float* p = out + (size_t)r * H2;
    float4 v = *reinterpret_cast<const float4*>(p + lane * 4);

    float mx = fmaxf(fmaxf(v.x, v.y), fmaxf(v.z, v.w));
    #pragma unroll
    for (int off = 16; off > 0; off >>= 1) mx = fmaxf(mx, __shfl_xor(mx, off, 32));
    float se = expf(v.x - mx) + expf(v.y - mx) + expf(v.z - mx) + expf(v.w - mx);
    #pragma unroll
    for (int off = 16; off > 0; off >>= 1) se += __shfl_xor(se, off, 32);
    float lse = mx + logf(se);

    v.x -= lse; v.y -= lse; v.z -= lse; v.w -= lse;
    *reinterpret_cast<float4*>(p + lane * 4) = v;
}

extern "C" void kernel_launch(void* const* d_in, const int* in_sizes, int n_in,
                              void* d_out, int out_size, void* d_ws, size_t ws_size,
                              hipStream_t stream) {
    const float* x  = (const float*)d_in[0];
    const int*   ei = (const int*)d_in[1];
    const float* W1 = (const float*)d_in[2];
    const float* b1 = (const float*)d_in[3];
    const float* W2 = (const float*)d_in[4];
    const float* b2 = (const float*)d_in[5];
    float* out = (float*)d_out;

    int N = in_sizes[0];          // x is [N,1]
    int E = in_sizes[1] / 2;      // edge_index is [2,E] row-major
    const int* row = ei;          // sources
    const int* col = ei + E;      // targets

    // Workspace: [dinv/deg: N][s: N][agg: N*64] -> 26.4 MB for N=100k
    float* dv  = (float*)d_ws;
    float* s   = dv + N;
    float* agg = s + N;

    const int B = 256;
    gcn_init_deg<<<(N + B - 1) / B, B, 0, stream>>>(dv, N);
    gcn_deg<<<(E + B - 1) / B, B, 0, stream>>>(col, dv, E);
    gcn_dinv_sinit<<<(N + B - 1) / B, B, 0, stream>>>(x, dv, s, N);
    gcn_sagg<<<4096, B, 0, stream>>>(x, row, col, dv, s, E);

    long long t3a = (long long)N * H1;
    gcn_agg_init<<<(unsigned)((t3a + B - 1) / B), B, 0, stream>>>(s, dv, W1, b1, agg, N);
    gcn_edge_agg<<<16384, B, 0, stream>>>(row, col, s, dv, W1, b1, agg, E);

    long long rowTiles = (N + 15) / 16;
    long long thr4 = rowTiles * 8 * 32;        // one wave per (rowTile, colTile)
    gcn_gemm_wmma<<<(unsigned)((thr4 + B - 1) / B), B, 0, stream>>>(agg, W2, b2, out, N);

    long long thr5 = (long long)N * 32;        // one wave per row
    gcn_log_softmax<<<(unsigned)((thr5 + B - 1) / B), B, 0, stream>>>(out, N);
}